// PathEmbeding2D_26242250179365
// MI455X (gfx1250) — compile-verified
//
#include <hip/hip_runtime.h>
#include <hip/hip_bf16.h>

typedef float v2f __attribute__((ext_vector_type(2)));
typedef float v4f __attribute__((ext_vector_type(4)));
typedef float v8f __attribute__((ext_vector_type(8)));

#define NH    16            // heads
#define HD    32            // headdim
#define LW    64            // H == W == 64
#define BATCH 16
#define NTILE (BATCH * LW * LW)   // 65536 (b, s) positions
#define NROW  (BATCH * LW * NH)   // 16384 rows per mask tensor

// ---------------------------------------------------------------------------
// Phase 1: dt = x @ W_dt^T via V_WMMA_F32_16X16X4_F32, then softplus * (-e^A).
// One wave handles one (b, s) position: A-matrix = 16 heads x K=32 slice of x,
// B-matrix = W_dt^T zero-padded to N=16 (branchless: clamped row + lane mask,
// so no EXEC-divergent regions interleave with the WMMA chain).
// ---------------------------------------------------------------------------
__global__ __launch_bounds__(256) void pe2d_dt_phase1(
    const float* __restrict__ x, const float* __restrict__ Wdt,
    const float* __restrict__ dt_bias, const float* __restrict__ A_log,
    float* __restrict__ alpha, float* __restrict__ beta)
{
  const int lane = threadIdx.x & 31;
  const int wave = threadIdx.x >> 5;
  const int tile = blockIdx.x * 8 + wave;       // 0 .. 65535
  const int b    = tile >> 12;
  const int s    = tile & 4095;
  const int Hi   = s >> 6;
  const int Wi   = s & 63;

  const float* xt = x + (size_t)tile * (NH * HD);  // 512 contiguous floats [h][d]

  // A-fragment (16x4 f32): lanes 0-15 hold M=lane, K={kb,kb+1};
  //                        lanes 16-31 hold M=lane-16, K={kb+2,kb+3}.
  const int M     = lane & 15;        // head row for A; N column for B
  const int khalf = (lane >> 4) * 2;  // 0 or 2

  // Branchless B fragments: B[K][N] = W_dt[N][K] for N<2, else 0.
  const float wmask = (M < 2) ? 1.f : 0.f;   // v_cndmask, no EXEC change
  const int   wrow  = (M < 2) ? M : 0;       // clamped, always a valid address
  v2f bfrag[8];
  v2f afrag[8];
#pragma unroll
  for (int step = 0; step < 8; ++step) {
    const int kb = step * 4 + khalf;
    v2f w = *(const v2f*)(Wdt + wrow * HD + kb);
    bfrag[step].x = w.x * wmask;
    bfrag[step].y = w.y * wmask;
    afrag[step] = __builtin_nontemporal_load((const v2f*)(xt + M * HD + kb));
  }

  v8f c = {};
#pragma unroll
  for (int step = 0; step < 8; ++step) {
    // (neg_a, A, neg_b, B, c_mod, C, reuse_a, reuse_b)
    c = __builtin_amdgcn_wmma_f32_16x16x4_f32(false, afrag[step], false,
                                              bfrag[step], (short)0, c,
                                              false, false);
  }

  // C/D 16x16 f32 layout: lane holds N=lane&15; VGPR v holds M=v (+8 for
  // lanes 16-31). Only N=0 (dt_alpha) and N=1 (dt_beta) columns are real.
  const int o = lane & 15;
  if (o < 2) {
    const int hbase = (lane < 16) ? 0 : 8;
#pragma unroll
    for (int v = 0; v < 8; ++v) {
      const int h = hbase + v;
      const float z  = c[v] + dt_bias[h];
      const float sp = (z > 20.f) ? z : log1pf(__expf(z));   // stable softplus
      const float val = sp * (-__expf(A_log[h]));
      if (o == 0)
        alpha[(((b * LW + Hi) * NH + h) * LW) + Wi] = val;   // (b,H,h,W)
      else
        beta[(((b * LW + Wi) * NH + h) * LW) + Hi] = val;    // (b,W,h,H)
    }
  }
}

// ---------------------------------------------------------------------------
// Phase 2: per 64-value row, cumsum (6-step Hillis-Steele scan in LDS) then
// emit the 64x64 segsum tile. out[i][j] = (i>=j) ? cs[i]-cs[j] : cs[j]-cs[i].
// 512 MB streamed out with coalesced non-temporal float4 stores.
// rid < NROW -> mask_w rows (alpha); rid >= NROW -> mask_h rows (beta),
// matching the concatenated (mask_w, mask_h) output layout exactly.
// ---------------------------------------------------------------------------
__global__ __launch_bounds__(256) void pe2d_mask_phase2(
    const float* __restrict__ alpha, const float* __restrict__ beta,
    float* __restrict__ out)
{
  __shared__ float cs[LW];

  const int rid = blockIdx.x;                 // 0 .. 2*NROW-1
  const int r   = rid & (NROW - 1);
  const float* in = (rid < NROW ? alpha : beta) + (size_t)r * LW;
  float* outp = out + (size_t)rid * (LW * LW);

  const int tid = threadIdx.x;
  if (tid < LW) cs[tid] = in[tid];
  __syncthreads();
#pragma unroll
  for (int ofs = 1; ofs < LW; ofs <<= 1) {    // inclusive prefix sum
    float t = (tid < LW && tid >= ofs) ? cs[tid - ofs] : 0.f;
    __syncthreads();
    if (tid < LW) cs[tid] += t;
    __syncthreads();
  }

#pragma unroll
  for (int k = 0; k < 4; ++k) {
    const int e4 = k * 256 + tid;             // float4 index, 0 .. 1023
    const int i  = e4 >> 4;                   // row
    const int j0 = (e4 & 15) * 4;             // first column of this float4
    const float csi = cs[i];
    v4f v;
#pragma unroll
    for (int m = 0; m < 4; ++m) {
      const int j = j0 + m;
      const float d = csi - cs[j];
      v[m] = (i >= j) ? d : -d;
    }
    __builtin_nontemporal_store(v, (v4f*)(outp + (size_t)e4 * 4));
  }
}

extern "C" void kernel_launch(void* const* d_in, const int* in_sizes, int n_in,
                              void* d_out, int out_size, void* d_ws, size_t ws_size,
                              hipStream_t stream) {
  const float* x       = (const float*)d_in[0];   // (16,64,64,512) f32
  const float* Wdt     = (const float*)d_in[1];   // (2,32) f32
  const float* dt_bias = (const float*)d_in[2];   // (16,) f32
  const float* A_log   = (const float*)d_in[3];   // (16,) f32
  float* out = (float*)d_out;                     // mask_w ++ mask_h, f32

  float* alpha = (float*)d_ws;                          // NROW*64 floats (4 MB)
  float* beta  = alpha + (size_t)NROW * LW;             // NROW*64 floats (4 MB)

  pe2d_dt_phase1<<<NTILE / 8, 256, 0, stream>>>(x, Wdt, dt_bias, A_log,
                                                alpha, beta);
  pe2d_mask_phase2<<<2 * NROW, 256, 0, stream>>>(alpha, beta, out);
}